// AttentionOrig_34471407518210
// MI455X (gfx1250) — compile-verified
//
#include <hip/hip_runtime.h>
#include <stdint.h>

#define BDIM 128
#define CDIM 49
#define NDIM 3136
#define CP   64   // padded C

typedef float v2f __attribute__((ext_vector_type(2)));
typedef float v8f __attribute__((ext_vector_type(8)));

__device__ __forceinline__ v8f wmma4(v2f a, v2f b, v8f c) {
    // D = A(16x4 f32) * B(4x16 f32) + C(16x16 f32)
    return __builtin_amdgcn_wmma_f32_16x16x4_f32(false, a, false, b, (short)0, c, false, false);
}

__device__ __forceinline__ v8f zero8() {
    v8f r = {0.f, 0.f, 0.f, 0.f, 0.f, 0.f, 0.f, 0.f};
    return r;
}

// CDNA5 async copy: global -> LDS, per-lane b128, tracked by ASYNCcnt.
__device__ __forceinline__ void async_b128(uint32_t ldsoff, const float* gaddr) {
    asm volatile("global_load_async_to_lds_b128 %0, %1, off"
                 :: "v"(ldsoff), "v"(gaddr) : "memory");
}

// ---------------------------------------------------------------------------
// Kernel 1: v[b,n] = mean over c of x[b,c,n]
// ---------------------------------------------------------------------------
__global__ void v_mean_kernel(const float* __restrict__ x, float* __restrict__ v) {
    int idx = blockIdx.x * blockDim.x + threadIdx.x;          // b*N + n, exact grid
    int b = idx / NDIM;
    int n = idx - b * NDIM;
    const float* xp = x + (size_t)b * CDIM * NDIM + n;
    float s = 0.f;
    #pragma unroll 7
    for (int c = 0; c < CDIM; ++c) s += xp[(size_t)c * NDIM];
    v[idx] = s * (1.0f / 49.0f);
}

// ---------------------------------------------------------------------------
// Kernel 2: q[b] = Wq_flat(49x3136) @ im2col(x[b])(3136x49), stride-8 8x8 conv.
// One block per batch, 4 waves; wave w owns M-tile w; K-outer loop reuses A
// across the 4 N-tiles. Output stored zero-padded to 64x64.
// ---------------------------------------------------------------------------
__global__ void __launch_bounds__(128) q_conv_kernel(const float* __restrict__ x,
                                                     const float* __restrict__ Wq,
                                                     float* __restrict__ qpad) {
    const int b    = blockIdx.x;
    const int wave = threadIdx.x >> 5;
    const int lane = threadIdx.x & 31;
    const int half = lane >> 4;
    const int l16  = lane & 15;
    const int mTile = wave;

    const int row   = mTile * 16 + l16;
    const float rmask = (row < CDIM) ? 1.f : 0.f;
    const int rowc  = (row < CDIM) ? row : 0;
    const float* xb = x + (size_t)b * CDIM * NDIM;
    const float* aRow = Wq + (size_t)rowc * (CDIM * 64);

    int   sbase[4];
    float pmask[4];
    #pragma unroll
    for (int nt = 0; nt < 4; ++nt) {
        int p  = nt * 16 + l16;
        int pc = (p < CDIM) ? p : 0;
        int oy = pc / 7, ox = pc - oy * 7;
        sbase[nt] = (oy * 8) * 56 + ox * 8;         // spatial origin of 8x8 window
        pmask[nt] = (p < CDIM) ? 1.f : 0.f;
    }

    v8f acc[4];
    #pragma unroll
    for (int nt = 0; nt < 4; ++nt) acc[nt] = zero8();

    for (int k = 0; k < CDIM * 64; k += 4) {
        const int kbase = k + half * 2;             // even -> 8B aligned pairs
        v2f a = *reinterpret_cast<const v2f*>(aRow + kbase);
        a.x *= rmask; a.y *= rmask;

        const int ci  = kbase >> 6;
        const int rem = kbase & 63;
        const int xoff = ci * NDIM + (rem >> 3) * 56 + (rem & 7);

        #pragma unroll
        for (int nt = 0; nt < 4; ++nt) {
            v2f bv = *reinterpret_cast<const v2f*>(xb + xoff + sbase[nt]);
            bv.x *= pmask[nt]; bv.y *= pmask[nt];
            acc[nt] = wmma4(a, bv, acc[nt]);
        }
    }

    float* qp = qpad + (size_t)b * CP * CP;
    #pragma unroll
    for (int r = 0; r < 8; ++r) {
        const int m = mTile * 16 + half * 8 + r;    // D layout: VGPR r -> M=r / M=r+8
        #pragma unroll
        for (int nt = 0; nt < 4; ++nt)
            qp[m * CP + nt * 16 + l16] = acc[nt][r];
    }
}

// ---------------------------------------------------------------------------
// Kernel 3: qW[b] = q[b] @ Wsr (49x49), qb[b] = q[b] @ bsr. Tiny VALU kernel.
// ---------------------------------------------------------------------------
__global__ void qw_kernel(const float* __restrict__ qpad, const float* __restrict__ Wsr,
                          const float* __restrict__ bsr, float* __restrict__ qWpad,
                          float* __restrict__ qb) {
    const int b = blockIdx.x;
    const float* qp = qpad + (size_t)b * CP * CP;
    for (int e = threadIdx.x; e < CP * CP; e += blockDim.x) {
        int c = e >> 6, ci = e & 63;
        float s = 0.f;
        if (c < CDIM && ci < CDIM) {
            for (int co = 0; co < CDIM; ++co)
                s += qp[c * CP + co] * Wsr[co * CDIM + ci];
        }
        qWpad[(size_t)b * CP * CP + e] = s;
    }
    if (threadIdx.x < CP) {
        int c = threadIdx.x;
        float s = 0.f;
        if (c < CDIM)
            for (int co = 0; co < CDIM; ++co) s += qp[c * CP + co] * bsr[co];
        qb[b * CP + c] = s;
    }
}

// ---------------------------------------------------------------------------
// Kernel 4: attn[b,c] = max_j ( qW[b](49x49) @ x[b](49x3136) )[c,j] + qb[b,c]
// Key-panels of x (padded 64 rows x 16 cols) are staged into LDS once per
// block via async global->LDS b128 copies (double-buffered, ASYNCcnt-
// pipelined), then consumed by all 4 waves -> 4x less x traffic.
// Every thread issues exactly 2 async b128 per panel so the per-wave
// ASYNCcnt discipline (wait <= 2 after issuing the next panel) is uniform.
// ---------------------------------------------------------------------------
__global__ void __launch_bounds__(128) score_max_kernel(const float* __restrict__ x,
                                                        const float* __restrict__ qWpad,
                                                        const float* __restrict__ qb,
                                                        float* __restrict__ attn) {
    __shared__ float sp[2][64 * 16];                // two 4KB panels
    const int b    = blockIdx.x;
    const int tid  = threadIdx.x;
    const int wave = tid >> 5;
    const int lane = tid & 31;
    const int half = lane >> 4;
    const int l16  = lane & 15;
    const int mTile = wave;

    const float* xb  = x + (size_t)b * CDIM * NDIM;
    const float* qWb = qWpad + (size_t)b * CP * CP;
    const int row = mTile * 16 + l16;               // qWpad zero-padded: always safe

    // Async-load slots: slot j in [0,256) covers panel floats [4j, 4j+4)
    // panel[r][c] = x[b, min(r,48), nt*16 + c]   (r in 0..63, c in 0..15)
    const int j0 = tid, j1 = tid + 128;
    const int r0 = j0 >> 2, q0 = j0 & 3;
    const int r1 = j1 >> 2, q1 = j1 & 3;
    const int rc0 = (r0 < CDIM) ? r0 : (CDIM - 1);
    const int rc1 = (r1 < CDIM) ? r1 : (CDIM - 1);
    const float* g0 = xb + (size_t)rc0 * NDIM + q0 * 4;
    const float* g1 = xb + (size_t)rc1 * NDIM + q1 * 4;
    const uint32_t lo0[2] = { (uint32_t)(size_t)&sp[0][j0 * 4],
                              (uint32_t)(size_t)&sp[1][j0 * 4] };
    const uint32_t lo1[2] = { (uint32_t)(size_t)&sp[0][j1 * 4],
                              (uint32_t)(size_t)&sp[1][j1 * 4] };

    // prologue: panel 0 -> buffer 0
    async_b128(lo0[0], g0);
    async_b128(lo1[0], g1);

    v2f a[13];
    #pragma unroll
    for (int kk = 0; kk < 13; ++kk)
        a[kk] = *reinterpret_cast<const v2f*>(qWb + row * CP + (kk * 4 + half * 2));

    v8f rmax = { -3.4e38f, -3.4e38f, -3.4e38f, -3.4e38f,
                 -3.4e38f, -3.4e38f, -3.4e38f, -3.4e38f };

    for (int nt = 0; nt < NDIM / 16; ++nt) {
        const int cur = nt & 1;
        if (nt < NDIM / 16 - 1) {
            async_b128(lo0[cur ^ 1], g0 + (nt + 1) * 16);
            async_b128(lo1[cur ^ 1], g1 + (nt + 1) * 16);
            // 2 newer ops in flight; <=2 means panel nt has fully landed
            asm volatile("s_wait_asynccnt 0x2" ::: "memory");
        } else {
            asm volatile("s_wait_asynccnt 0x0" ::: "memory");
        }
        __syncthreads();                            // whole panel visible to block

        const float* p = &sp[cur][0];
        v8f acc = zero8();
        #pragma unroll
        for (int kk = 0; kk < 13; ++kk) {
            const int   kb = kk * 4 + half * 2;     // <= 51 < 64: in-panel
            const float m0 = (kb     < CDIM) ? 1.f : 0.f;
            const float m1 = (kb + 1 < CDIM) ? 1.f : 0.f;
            v2f bv;
            bv.x = p[kb * 16 + l16] * m0;
            bv.y = p[(kb + 1) * 16 + l16] * m1;
            acc = wmma4(a[kk], bv, acc);
        }
        #pragma unroll
        for (int r = 0; r < 8; ++r) rmax[r] = fmaxf(rmax[r], acc[r]);

        __syncthreads();                            // done reading before overwrite
    }

    #pragma unroll
    for (int m = 1; m < 16; m <<= 1) {
        #pragma unroll
        for (int r = 0; r < 8; ++r) {
            float o = __shfl_xor(rmax[r], m, 32);
            rmax[r] = fmaxf(rmax[r], o);
        }
    }
    if (l16 == 0) {
        #pragma unroll
        for (int r = 0; r < 8; ++r) {
            const int rowD = mTile * 16 + half * 8 + r;
            attn[b * CP + rowD] = rmax[r] + qb[b * CP + rowD];
        }
    }
}

// ---------------------------------------------------------------------------
// Kernel 5: pv(3136x128) = Wproj(3136x3136) @ v^T(3136x128). Each wave owns
// one M-tile x four 16-wide B-tiles: one A load feeds 4 WMMAs, cutting Wproj
// re-reads from 8x to 2x. 392 waves, all operands coalesced 8B loads.
// ---------------------------------------------------------------------------
__global__ void __launch_bounds__(128) pv_gemm_kernel(const float* __restrict__ Wproj,
                                                      const float* __restrict__ v,
                                                      float* __restrict__ pv) {
    const int gw   = blockIdx.x * 4 + (threadIdx.x >> 5);
    const int lane = threadIdx.x & 31;
    const int half = lane >> 4;
    const int l16  = lane & 15;
    const int mTile = gw >> 1;       // 196 tiles over M=3136
    const int g     = gw & 1;        // which 64-wide half of B=128

    const int row = mTile * 16 + l16;
    const float* ap = Wproj + (size_t)row * NDIM + half * 2;
    const float* bp[4];
    #pragma unroll
    for (int t = 0; t < 4; ++t)
        bp[t] = v + (size_t)(g * 64 + t * 16 + l16) * NDIM + half * 2;

    v8f acc[4];
    #pragma unroll
    for (int t = 0; t < 4; ++t) acc[t] = zero8();

    for (int k = 0; k < NDIM; k += 4) {
        __builtin_prefetch(ap + k + 256, 0, 1);
        v2f a = *reinterpret_cast<const v2f*>(ap + k);
        #pragma unroll
        for (int t = 0; t < 4; ++t) {
            v2f bv = *reinterpret_cast<const v2f*>(bp[t] + k);
            acc[t] = wmma4(a, bv, acc[t]);
        }
    }

    #pragma unroll
    for (int r = 0; r < 8; ++r) {
        const int m = mTile * 16 + half * 8 + r;
        #pragma unroll
        for (int t = 0; t < 4; ++t)
            pv[(size_t)m * BDIM + g * 64 + t * 16 + l16] = acc[t][r];
    }
}

// ---------------------------------------------------------------------------
// Kernel 6: final[b,m,c] = pv[m,b] * attn[b,c]   (output layout (B, N, C))
// ---------------------------------------------------------------------------
__global__ void outer_kernel(const float* __restrict__ pv, const float* __restrict__ attn,
                             float* __restrict__ out) {
    const int total = BDIM * NDIM * CDIM;
    int i = blockIdx.x * blockDim.x + threadIdx.x;
    if (i >= total) return;
    int c = i % CDIM;
    int t = i / CDIM;
    int m = t % NDIM;
    int b = t / NDIM;
    out[i] = pv[(size_t)m * BDIM + b] * attn[b * CP + c];
}

// ---------------------------------------------------------------------------
extern "C" void kernel_launch(void* const* d_in, const int* in_sizes, int n_in,
                              void* d_out, int out_size, void* d_ws, size_t ws_size,
                              hipStream_t stream) {
    const float* x     = (const float*)d_in[0];   // (B, C, N)
    const float* Wq    = (const float*)d_in[1];   // (C, C, 8, 8)
    const float* Wsr   = (const float*)d_in[2];   // (C, C)
    const float* bsr   = (const float*)d_in[3];   // (C,)
    const float* Wproj = (const float*)d_in[4];   // (N, N)
    float* out = (float*)d_out;                   // (B, N, C)

    float* ws    = (float*)d_ws;
    float* vbuf  = ws;                            // B*N      = 401408
    float* qpad  = vbuf + BDIM * NDIM;            // B*64*64  = 524288
    float* qWp   = qpad + BDIM * CP * CP;         // B*64*64  = 524288
    float* qbv   = qWp  + BDIM * CP * CP;         // B*64     = 8192
    float* attn  = qbv  + BDIM * CP;              // B*64     = 8192
    float* pv    = attn + BDIM * CP;              // N*B      = 401408
    (void)in_sizes; (void)n_in; (void)out_size; (void)ws_size;

    v_mean_kernel   <<<(BDIM * NDIM) / 256, 256, 0, stream>>>(x, vbuf);
    q_conv_kernel   <<<BDIM, 128, 0, stream>>>(x, Wq, qpad);
    qw_kernel       <<<BDIM, 256, 0, stream>>>(qpad, Wsr, bsr, qWp, qbv);
    score_max_kernel<<<BDIM, 128, 0, stream>>>(x, qWp, qbv, attn);
    pv_gemm_kernel  <<<(NDIM / 16) * 2 / 4, 128, 0, stream>>>(Wproj, vbuf, pv);
    outer_kernel    <<<(BDIM * NDIM * CDIM + 255) / 256, 256, 0, stream>>>(pv, attn, out);
}